// SCLayer_3341484556724
// MI455X (gfx1250) — compile-verified
//
#include <hip/hip_runtime.h>
#include <math.h>

// Problem constants (match reference)
#define Bdim 4096
#define Mdim 306
#define Ndim 984
#define MpD  320      // padded M (multiple of 32)
#define NpD  992      // padded N (multiple of 32)
#define ITERS 50
#define PITERS 128
#define LAMBDA_F 0.1f

typedef float v2f __attribute__((ext_vector_type(2)));
typedef float v8f __attribute__((ext_vector_type(8)));

__device__ __forceinline__ v8f wmma4(v2f a, v2f b, v8f c) {
    // D = A(16x4 f32) * B(4x16 f32) + C(16x16 f32)
    return __builtin_amdgcn_wmma_f32_16x16x4_f32(false, a, false, b, (short)0, c,
                                                 false, false);
}

// ---------------------------------------------------------------------------
// Zero-padded phi [MpD x NpD] and transposed phi [NpD x MpD]: GEMM inner
// loops and epilogues become fully branch-free.
// ---------------------------------------------------------------------------
__global__ __launch_bounds__(256) void pad_phi_kernel(const float* __restrict__ pr,
                                                      const float* __restrict__ pi,
                                                      float* __restrict__ pPr,
                                                      float* __restrict__ pPi,
                                                      float* __restrict__ pTr,
                                                      float* __restrict__ pTi) {
    int i = blockIdx.x * 256 + threadIdx.x;
    const int total = MpD * NpD;
    if (i >= total) return;
    int m = i / NpD;
    int n = i - m * NpD;
    float vr = 0.0f, vi = 0.0f;
    if (m < Mdim && n < Ndim) {
        vr = pr[(size_t)m * Ndim + n];
        vi = pi[(size_t)m * Ndim + n];
    }
    pPr[(size_t)m * NpD + n] = vr;
    pPi[(size_t)m * NpD + n] = vi;
    pTr[(size_t)n * MpD + m] = vr;
    pTi[(size_t)n * MpD + m] = vi;
}

// ---------------------------------------------------------------------------
// Zero-padded r: [B x MpD] planes.
// ---------------------------------------------------------------------------
__global__ __launch_bounds__(256) void pad_r_kernel(const float* __restrict__ rr,
                                                    const float* __restrict__ ri,
                                                    float* __restrict__ rPr,
                                                    float* __restrict__ rPi) {
    size_t i = (size_t)blockIdx.x * 256 + threadIdx.x;
    const size_t total = (size_t)Bdim * MpD;
    if (i >= total) return;
    int b = (int)(i / MpD);
    int m = (int)(i - (size_t)b * MpD);
    float vr = 0.0f, vi = 0.0f;
    if (m < Mdim) {
        vr = rr[(size_t)b * Mdim + m];
        vi = ri[(size_t)b * Mdim + m];
    }
    rPr[i] = vr;
    rPi[i] = vi;
}

// ---------------------------------------------------------------------------
// Power iteration on v -> phi^H (phi v) to get lambda = sigma_max(phi)^2.
// Single workgroup of 1024 threads; deterministic init v = 1/sqrt(N).
// ---------------------------------------------------------------------------
__global__ __launch_bounds__(1024) void power_kernel(const float* __restrict__ pr,
                                                     const float* __restrict__ pi,
                                                     float* vr, float* vi,
                                                     float* ur, float* ui,
                                                     float* sc) {
    __shared__ float red[1024];
    const int tid = threadIdx.x;
    const float v0 = 1.0f / sqrtf((float)Ndim);
    for (int n = tid; n < Ndim; n += 1024) { vr[n] = v0; vi[n] = 0.0f; }
    __syncthreads();

    float lambda = 1.0f;
    for (int it = 0; it < PITERS; ++it) {
        // u = phi * v   (M x 1)
        for (int m = tid; m < Mdim; m += 1024) {
            const float* prow = pr + (size_t)m * Ndim;
            const float* irow = pi + (size_t)m * Ndim;
            float sr = 0.0f, si = 0.0f;
            for (int n = 0; n < Ndim; ++n) {
                float a = prow[n], b = irow[n], c = vr[n], d = vi[n];
                sr += a * c - b * d;
                si += a * d + b * c;
            }
            ur[m] = sr; ui[m] = si;
        }
        __syncthreads();
        // v' = phi^H * u  (N x 1), accumulate |v'|^2
        float part = 0.0f;
        for (int n = tid; n < Ndim; n += 1024) {
            float sr = 0.0f, si = 0.0f;
            for (int m = 0; m < Mdim; ++m) {
                float a = pr[(size_t)m * Ndim + n];
                float b = pi[(size_t)m * Ndim + n];
                float c = ur[m], d = ui[m];
                sr += a * c + b * d;   // conj(phi) * u, real
                si += a * d - b * c;   // imag
            }
            vr[n] = sr; vi[n] = si;
            part += sr * sr + si * si;
        }
        red[tid] = part;
        __syncthreads();
        for (int s = 512; s > 0; s >>= 1) {
            if (tid < s) red[tid] += red[tid + s];
            __syncthreads();
        }
        lambda = sqrtf(red[0]);           // ||G v|| -> sigma_max^2 as v converges
        float invl = 1.0f / lambda;
        for (int n = tid; n < Ndim; n += 1024) { vr[n] *= invl; vi[n] *= invl; }
        __syncthreads();
    }
    if (tid == 0) {
        float step = 1.0f / lambda;       // 1 / ||phi||_2^2
        sc[0] = step;
        sc[1] = LAMBDA_F * step;
    }
}

// ---------------------------------------------------------------------------
// error[b,m] = sum_n w[b,n]*phi[m,n] - r[b,m]   (complex; 4 real WMMA accs per
// 16x16 tile).  One wave computes a 16(b) x 32(m) tile (two col-tiles sharing
// the A fragment).  Fully branch-free: all operands padded.
// ---------------------------------------------------------------------------
__global__ __launch_bounds__(256) void err_kernel(const float* __restrict__ wr,
                                                  const float* __restrict__ wi,
                                                  const float* __restrict__ pPr,
                                                  const float* __restrict__ pPi,
                                                  const float* __restrict__ rPr,
                                                  const float* __restrict__ rPi,
                                                  float* __restrict__ er,
                                                  float* __restrict__ ei) {
    const int lane = threadIdx.x & 31;
    const int wave = threadIdx.x >> 5;
    const int b0 = (blockIdx.x * 8 + wave) * 16;
    const int m0 = blockIdx.y * 32;
    const int hl = lane >> 4;          // K half-select (0 -> K{0,1}, 1 -> K{2,3})
    const int ll = lane & 15;

    const float* wrp = wr + (size_t)(b0 + ll) * NpD;        // A row (b), padded w
    const float* wip = wi + (size_t)(b0 + ll) * NpD;
    const float* b0r = pPr + (size_t)(m0 + ll) * NpD;       // B col-tile 0 (m)
    const float* b0i = pPi + (size_t)(m0 + ll) * NpD;
    const float* b1r = pPr + (size_t)(m0 + 16 + ll) * NpD;  // B col-tile 1
    const float* b1i = pPi + (size_t)(m0 + 16 + ll) * NpD;

    v8f aRR0 = {}, aII0 = {}, aRI0 = {}, aIR0 = {};
    v8f aRR1 = {}, aII1 = {}, aRI1 = {}, aIR1 = {};
#pragma unroll 2
    for (int k = 0; k < Ndim; k += 4) {
        const int kb = k + hl * 2;
        v2f ar  = *(const v2f*)(wrp + kb);
        v2f ai  = *(const v2f*)(wip + kb);
        v2f br0 = *(const v2f*)(b0r + kb);
        v2f bi0 = *(const v2f*)(b0i + kb);
        v2f br1 = *(const v2f*)(b1r + kb);
        v2f bi1 = *(const v2f*)(b1i + kb);
        aRR0 = wmma4(ar, br0, aRR0);
        aII0 = wmma4(ai, bi0, aII0);
        aRI0 = wmma4(ar, bi0, aRI0);
        aIR0 = wmma4(ai, br0, aIR0);
        aRR1 = wmma4(ar, br1, aRR1);
        aII1 = wmma4(ai, bi1, aII1);
        aRI1 = wmma4(ar, bi1, aRI1);
        aIR1 = wmma4(ai, br1, aIR1);
    }
    // C/D layout: VGPR j -> row (b0 + j + 8*hl), col (tile_base + ll)
#pragma unroll
    for (int t = 0; t < 2; ++t) {
        const int col = m0 + t * 16 + ll;
        const v8f* cRR = t ? &aRR1 : &aRR0;
        const v8f* cII = t ? &aII1 : &aII0;
        const v8f* cRI = t ? &aRI1 : &aRI0;
        const v8f* cIR = t ? &aIR1 : &aIR0;
#pragma unroll
        for (int j = 0; j < 8; ++j) {
            const int row = b0 + j + hl * 8;
            const size_t idx = (size_t)row * MpD + col;
            er[idx] = (*cRR)[j] - (*cII)[j] - rPr[idx];
            ei[idx] = (*cRI)[j] + (*cIR)[j] - rPi[idx];
        }
    }
}

// ---------------------------------------------------------------------------
// grad[b,n] = sum_m error[b,m]*conj(phi[m,n]);  w = shrink(w - step*grad).
// One wave computes a 16(b) x 32(n) tile.  B fragments from padded TRANSPOSED
// phi (contiguous float2).  Fully branch-free; w pad columns stay exactly 0.
// ---------------------------------------------------------------------------
__global__ __launch_bounds__(256) void grad_kernel(const float* __restrict__ er,
                                                   const float* __restrict__ ei,
                                                   const float* __restrict__ pTr,
                                                   const float* __restrict__ pTi,
                                                   float* __restrict__ wr,
                                                   float* __restrict__ wi,
                                                   const float* __restrict__ sc) {
    const int lane = threadIdx.x & 31;
    const int wave = threadIdx.x >> 5;
    const int b0 = (blockIdx.x * 8 + wave) * 16;
    const int n0 = blockIdx.y * 32;
    const int hl = lane >> 4;
    const int ll = lane & 15;

    const float step = sc[0];
    const float thresh = sc[1];

    const float* erp = er + (size_t)(b0 + ll) * MpD;        // A row (b)
    const float* eip = ei + (size_t)(b0 + ll) * MpD;
    const float* b0r = pTr + (size_t)(n0 + ll) * MpD;       // B col-tile 0 (n)
    const float* b0i = pTi + (size_t)(n0 + ll) * MpD;
    const float* b1r = pTr + (size_t)(n0 + 16 + ll) * MpD;  // B col-tile 1
    const float* b1i = pTi + (size_t)(n0 + 16 + ll) * MpD;

    v8f aRR0 = {}, aII0 = {}, aRI0 = {}, aIR0 = {};
    v8f aRR1 = {}, aII1 = {}, aRI1 = {}, aIR1 = {};
#pragma unroll 2
    for (int k = 0; k < MpD; k += 4) {
        const int kb = k + hl * 2;
        v2f ar  = *(const v2f*)(erp + kb);   // error (pad cols exact zero)
        v2f ai  = *(const v2f*)(eip + kb);
        v2f br0 = *(const v2f*)(b0r + kb);   // phiT (pads zero)
        v2f bi0 = *(const v2f*)(b0i + kb);
        v2f br1 = *(const v2f*)(b1r + kb);
        v2f bi1 = *(const v2f*)(b1i + kb);
        aRR0 = wmma4(ar, br0, aRR0);  // sum er*pr
        aII0 = wmma4(ai, bi0, aII0);  // sum ei*pi
        aRI0 = wmma4(ar, bi0, aRI0);  // sum er*pi
        aIR0 = wmma4(ai, br0, aIR0);  // sum ei*pr
        aRR1 = wmma4(ar, br1, aRR1);
        aII1 = wmma4(ai, bi1, aII1);
        aRI1 = wmma4(ar, bi1, aRI1);
        aIR1 = wmma4(ai, br1, aIR1);
    }
    // grad = (er*pr + ei*pi) + i(ei*pr - er*pi);  proximal (soft-mag) update
#pragma unroll
    for (int t = 0; t < 2; ++t) {
        const int col = n0 + t * 16 + ll;
        const v8f* cRR = t ? &aRR1 : &aRR0;
        const v8f* cII = t ? &aII1 : &aII0;
        const v8f* cRI = t ? &aRI1 : &aRI0;
        const v8f* cIR = t ? &aIR1 : &aIR0;
#pragma unroll
        for (int j = 0; j < 8; ++j) {
            const int row = b0 + j + hl * 8;
            const size_t idx = (size_t)row * NpD + col;
            float gr = (*cRR)[j] + (*cII)[j];
            float gi = (*cIR)[j] - (*cRI)[j];
            float nr = wr[idx] - step * gr;
            float ni = wi[idx] - step * gi;
            float mag = sqrtf(nr * nr + ni * ni);
            float s = (mag > thresh) ? (1.0f - thresh / mag) : 0.0f;
            wr[idx] = s * nr;
            wi[idx] = s * ni;
        }
    }
}

// ---------------------------------------------------------------------------
// out[b,n] = |w[b,n]| , reading padded w planes, writing compact [B x N].
// ---------------------------------------------------------------------------
__global__ __launch_bounds__(256) void abs_kernel(const float* __restrict__ wr,
                                                  const float* __restrict__ wi,
                                                  float* __restrict__ out) {
    size_t i = (size_t)blockIdx.x * blockDim.x + threadIdx.x;
    const size_t total = (size_t)Bdim * Ndim;
    if (i < total) {
        int b = (int)(i / Ndim);
        int n = (int)(i - (size_t)b * Ndim);
        size_t idx = (size_t)b * NpD + n;
        float a = wr[idx], c = wi[idx];
        out[i] = sqrtf(a * a + c * c);
    }
}

extern "C" void kernel_launch(void* const* d_in, const int* in_sizes, int n_in,
                              void* d_out, int out_size, void* d_ws, size_t ws_size,
                              hipStream_t stream) {
    const float* rr = (const float*)d_in[0];  // r_real  [B,M]
    const float* ri = (const float*)d_in[1];  // r_imag  [B,M]
    const float* pr = (const float*)d_in[2];  // phi_real [M,N]
    const float* pi = (const float*)d_in[3];  // phi_imag [M,N]

    float* ws = (float*)d_ws;
    const size_t BNp = (size_t)Bdim * NpD;
    const size_t BMp = (size_t)Bdim * MpD;
    const size_t PP  = (size_t)MpD * NpD;
    float* wr  = ws;              // w real  [B x NpD] (padded)
    float* wi  = wr + BNp;        // w imag
    float* er  = wi + BNp;        // error real [B x MpD] (padded)
    float* ei  = er + BMp;        // error imag
    float* rPr = ei + BMp;        // padded r real [B x MpD]
    float* rPi = rPr + BMp;       // padded r imag
    float* pPr = rPi + BMp;       // padded phi   [MpD x NpD]
    float* pPi = pPr + PP;
    float* pTr = pPi + PP;        // padded phi^T [NpD x MpD]
    float* pTi = pTr + PP;
    float* vr  = pTi + PP;        // power-iter vectors
    float* vi  = vr + Ndim;
    float* ur  = vi + Ndim;
    float* ui  = ur + Mdim;
    float* sc  = ui + Mdim;       // {step, thresh}

    // w0 = 0 (pad columns must also start 0 and will remain 0)
    hipMemsetAsync(wr, 0, 2 * BNp * sizeof(float), stream);

    // One-time padded operands
    pad_phi_kernel<<<(unsigned)((PP + 255) / 256), 256, 0, stream>>>(pr, pi, pPr,
                                                                     pPi, pTr, pTi);
    pad_r_kernel<<<(unsigned)((BMp + 255) / 256), 256, 0, stream>>>(rr, ri, rPr,
                                                                    rPi);

    // step = 1/||phi||_2^2, thresh = lambda*step (computed on device)
    power_kernel<<<1, 1024, 0, stream>>>(pr, pi, vr, vi, ur, ui, sc);

    dim3 gA(Bdim / 128, MpD / 32);  // (32, 10): 8 waves/block, 16x32 tile/wave
    dim3 gB(Bdim / 128, NpD / 32);  // (32, 31)
    for (int it = 0; it < ITERS; ++it) {
        err_kernel<<<gA, 256, 0, stream>>>(wr, wi, pPr, pPi, rPr, rPi, er, ei);
        grad_kernel<<<gB, 256, 0, stream>>>(er, ei, pTr, pTi, wr, wi, sc);
    }

    abs_kernel<<<(unsigned)(((size_t)Bdim * Ndim + 255) / 256), 256, 0, stream>>>(
        wr, wi, (float*)d_out);
}